// CRF_84937273246139
// MI455X (gfx1250) — compile-verified
//
#include <hip/hip_runtime.h>
#include <stdint.h>

#define Bb 64
#define Tt 512
#define Dd 1024
#define Uu 128

typedef float v2f __attribute__((ext_vector_type(2)));
typedef float v8f __attribute__((ext_vector_type(8)));

// ---------------------------------------------------------------------------
// Kernel A: potentials = inputs @ kernel + bias + boundary energies
// M = B*T = 32768 rows, K = D = 1024, N = U = 128.
// One block = 16 rows x 128 cols; 8 waves, each wave one 16x16 tile via
// V_WMMA_F32_16X16X4_F32 (full fp32, matches reference numerics).
// ---------------------------------------------------------------------------
__global__ __launch_bounds__(256) void crf_potentials_wmma(
    const float* __restrict__ A,       // (B*T, D) row-major
    const float* __restrict__ Bm,      // (D, U)   row-major
    const float* __restrict__ bias,    // (U,)
    const uint8_t* __restrict__ mask,  // (B*T,) bool bytes
    const float* __restrict__ leftb,   // (U,)
    const float* __restrict__ rightb,  // (U,)
    float* __restrict__ out)           // (B*T, U)
{
  const int lane = threadIdx.x & 31;
  const int wave = threadIdx.x >> 5;
  const int m0 = blockIdx.x * 16;      // M tile
  const int n0 = wave * 16;            // N tile (8 waves cover N=128)
  const int row = lane & 15;
  const int grp = lane >> 4;           // 0: K pair {k,k+1}, 1: {k+2,k+3}
  const int n = n0 + row;

  // A layout (16x4 f32): lane<16 -> M=lane, VGPR0=K0, VGPR1=K1; lane>=16 -> K2,K3
  const float* ap = A + (size_t)(m0 + row) * Dd + 2 * grp;
  // B layout (4x16 f32): VGPR0: lanes0-15 K=0 / lanes16-31 K=2; VGPR1: K=1 / K=3
  const float* bpk = Bm + (size_t)(2 * grp) * Uu + n;

  v8f acc = {};
#pragma unroll 4
  for (int k = 0; k < Dd; k += 4) {
    v2f a;
    a.x = ap[0];
    a.y = ap[1];
    v2f b;
    b.x = bpk[0];
    b.y = bpk[Uu];
    acc = __builtin_amdgcn_wmma_f32_16x16x4_f32(
        /*neg_a=*/false, a, /*neg_b=*/false, b,
        /*c_mod=*/(short)0, acc, /*reuse_a=*/false, /*reuse_b=*/false);
    ap += 4;
    bpk += (size_t)4 * Uu;
  }

  const float bn = bias[n];
  const float lv = leftb[n];
  const float rv = rightb[n];

  // C/D layout: VGPR r, lanes 0-15 -> M=r, lanes 16-31 -> M=r+8; N = lane%16
#pragma unroll
  for (int r = 0; r < 8; ++r) {
    const int m = m0 + r + 8 * grp;    // global row, never crosses batch (T%16==0)
    const int t = m & (Tt - 1);
    const uint32_t mb = mask[m];
    const uint32_t mprev = (t > 0) ? (uint32_t)mask[m - 1] : 0u;
    const uint32_t mnext = (t < Tt - 1) ? (uint32_t)mask[m + 1] : 0u;
    float v = acc[r] + bn;
    if (mb && !mprev) v += lv;         // start-of-run boundary
    if (mb && !mnext) v += rv;         // end-of-run boundary
    out[(size_t)m * Uu + n] = v;
  }
}

// ---------------------------------------------------------------------------
// Kernel B: Viterbi forward scan, MAX-ONLY (argmax recovered in backtrack).
// One block per batch, one thread per tag j. Each thread keeps its chain
// column (128 floats) in VGPRs; state[128] broadcast from LDS (ds_load_b128
// merged). Masked steps skip the inner loop (block-uniform branch).
// Stores the state row after every step (states[b][t][:]) for the backtrack.
// ---------------------------------------------------------------------------
__global__ __launch_bounds__(128) void crf_viterbi_fwd(
    const float* __restrict__ pot,     // (B,T,U) from kernel A
    const uint8_t* __restrict__ mask,  // (B,T)
    const float* __restrict__ chain,   // (U,U)
    float* __restrict__ states,        // (B,T,U) ws: state after step t
    int* __restrict__ last_tag)        // (B,) ws
{
  __shared__ float s_state[Uu];
  __shared__ uint8_t s_mask[Tt];

  const int b = blockIdx.x;
  const int j = threadIdx.x;

  // chain column j resident in VGPRs (constant indices after full unroll)
  float cj[Uu];
#pragma unroll
  for (int i = 0; i < Uu; ++i) cj[i] = chain[i * Uu + j];

#pragma unroll
  for (int k = 0; k < Tt / Uu; ++k) s_mask[j + k * Uu] = mask[b * Tt + j + k * Uu];

  float myst = pot[((size_t)b * Tt) * Uu + j];
  s_state[j] = myst;
  states[((size_t)b * Tt) * Uu + j] = myst;  // row 0 = init
  __syncthreads();

  for (int t = 1; t < Tt; ++t) {
    const size_t base = ((size_t)b * Tt + t) * Uu;
    const bool m = s_mask[t] != 0;     // uniform across block
    if (m) {
      float best = -3.402823466e38f;
#pragma unroll
      for (int i = 0; i < Uu; ++i) best = fmaxf(best, s_state[i] + cj[i]);
      const float p = pot[base + j];
      __syncthreads();                  // all reads of s_state done
      myst = p + best;
      s_state[j] = myst;
      __syncthreads();
    }
    states[base + j] = myst;            // state entering step t+1
  }

  if (j == 0) {                         // last_tag = argmax(final_state), first-max
    float best = s_state[0];
    int arg = 0;
    for (int i = 1; i < Uu; ++i) {
      if (s_state[i] > best) { best = s_state[i]; arg = i; }
    }
    last_tag[b] = arg;
  }
}

// ---------------------------------------------------------------------------
// Kernel C: backtrack with on-the-fly backpointer recompute + one-hot.
// Per batch: chain^T (64KB) + a 128-row state chunk (64KB) staged in LDS
// (130KB of the 320KB WGP LDS). Each of the 511 sequential steps is one
// 128-wide add plus a shuffle/LDS argmax-first reduction (bit-identical
// values to the forward pass, first-max tie-breaking preserved).
// ---------------------------------------------------------------------------
__global__ __launch_bounds__(128) void crf_backtrack(
    const float* __restrict__ states,  // (B,T,U)
    const uint8_t* __restrict__ mask,  // (B,T)
    const float* __restrict__ chain,   // (U,U)
    const int* __restrict__ last_tag,  // (B,)
    float* __restrict__ onehot)        // (B,T,U)
{
  extern __shared__ unsigned char dynsmem[];
  float* s_chainT = (float*)dynsmem;              // U*U floats = 64 KB (chainT[j][i])
  float* s_chunk = s_chainT + Uu * Uu;            // 128 rows x U floats = 64 KB
  int* s_tag = (int*)(s_chunk + 128 * Uu);        // T ints
  float* s_redv = (float*)(s_tag + Tt);           // 4 floats
  int* s_redi = (int*)(s_redv + 4);               // 4 ints
  int* s_cur = s_redi + 4;                        // 1 int
  uint8_t* s_msk = (uint8_t*)(s_cur + 1);         // T bytes

  const int b = blockIdx.x;
  const int tid = threadIdx.x;
  const int w = tid >> 5;
  const int lane = tid & 31;

  // stage chain transposed: s_chainT[j*U + i] = chain[i*U + j] (coalesced reads)
  for (int idx = tid; idx < Uu * Uu; idx += 128) {
    const int i = idx >> 7;
    const int jj = idx & (Uu - 1);
    s_chainT[jj * Uu + i] = chain[idx];
  }
  for (int idx = tid; idx < Tt; idx += 128) s_msk[idx] = mask[b * Tt + idx];

  int tag = last_tag[b];               // uniform
  if (tid == 0) s_tag[Tt - 1] = tag;
  const float* st_b = states + (size_t)b * Tt * Uu;

  for (int c = 3; c >= 0; --c) {
    const int row_base = c * 128;
    __syncthreads();                   // previous chunk reads done / staging done
    const float4* src = (const float4*)(st_b + (size_t)row_base * Uu);
    float4* dst = (float4*)s_chunk;
    for (int idx = tid; idx < 128 * Uu / 4; idx += 128) dst[idx] = src[idx];
    __syncthreads();

    const int t_hi = (c == 3) ? (Tt - 1) : (row_base + 128);
    for (int t = t_hi; t >= row_base + 1; --t) {
      if (s_msk[t]) {                  // uniform branch; masked step -> identity bp
        const float v = s_chunk[(t - 1 - row_base) * Uu + tid] + s_chainT[tag * Uu + tid];
        float bv = v;
        int bi = tid;
#pragma unroll
        for (int off = 16; off >= 1; off >>= 1) {
          const float ov = __shfl_xor(bv, off, 32);
          const int oi = __shfl_xor(bi, off, 32);
          if (ov > bv || (ov == bv && oi < bi)) { bv = ov; bi = oi; }
        }
        if (lane == 0) { s_redv[w] = bv; s_redi[w] = bi; }
        __syncthreads();
        if (tid == 0) {
          float gv = s_redv[0];
          int gi = s_redi[0];
#pragma unroll
          for (int k = 1; k < 4; ++k) {
            if (s_redv[k] > gv || (s_redv[k] == gv && s_redi[k] < gi)) {
              gv = s_redv[k];
              gi = s_redi[k];
            }
          }
          *s_cur = gi;
        }
        __syncthreads();
        tag = *s_cur;
      }
      if (tid == 0) s_tag[t - 1] = tag;
    }
  }
  __syncthreads();

  for (int t = 0; t < Tt; ++t) {
    onehot[((size_t)b * Tt + t) * Uu + tid] = (tid == s_tag[t]) ? 1.0f : 0.0f;
  }
}

// ---------------------------------------------------------------------------
extern "C" void kernel_launch(void* const* d_in, const int* in_sizes, int n_in,
                              void* d_out, int out_size, void* d_ws, size_t ws_size,
                              hipStream_t stream) {
  const float* inputs = (const float*)d_in[0];    // (B,T,D) f32
  const uint8_t* mask = (const uint8_t*)d_in[1];  // (B,T) bool bytes
  const float* kern = (const float*)d_in[2];      // (D,U) f32
  const float* bias = (const float*)d_in[3];      // (U,) f32
  const float* chain = (const float*)d_in[4];     // (U,U) f32
  const float* lb = (const float*)d_in[5];        // (U,) f32
  const float* rb = (const float*)d_in[6];        // (U,) f32

  float* potentials = (float*)d_out;                      // first output
  float* onehot = potentials + (size_t)Bb * Tt * Uu;      // second output

  float* states = (float*)d_ws;                           // B*T*U floats
  int* last_tag = (int*)((float*)d_ws + (size_t)Bb * Tt * Uu);

  // Phase 1: WMMA GEMM + bias + boundary energies
  crf_potentials_wmma<<<(Bb * Tt) / 16, 256, 0, stream>>>(
      inputs, kern, bias, mask, lb, rb, potentials);

  // Phase 2: Viterbi forward scan, max-only (64 blocks, one per batch)
  crf_viterbi_fwd<<<Bb, 128, 0, stream>>>(potentials, mask, chain, states, last_tag);

  // Phase 3: backtrack (recompute argmax along taken path) + one-hot
  const size_t smemC = (size_t)(Uu * Uu + 128 * Uu) * sizeof(float) +
                       (size_t)Tt * sizeof(int) + 9 * sizeof(int) + Tt;
  crf_backtrack<<<Bb, 128, smemC, stream>>>(states, mask, chain, last_tag, onehot);
}